// SE3EquivariantTransformerMixedHeads_14834817040539
// MI455X (gfx1250) — compile-verified
//
#include <hip/hip_runtime.h>

// ---------------- problem constants ----------------
constexpr int kN   = 10000;   // nodes
constexpr int kE   = 160000;  // edges
constexpr int kC   = 16;      // channels
constexpr int kC2  = 8;
constexpr int kHid = 64;
constexpr int kNH  = 2;
constexpr int kNG  = 16;
constexpr int kFin = 64;
constexpr int kFout= 32;

// ---------------- workspace layout (float units) ----------------
constexpr int OFF_X     = 0;                          // [N, C]
constexpr int OFF_Q     = OFF_X   + kN * kC;          // [NH, N, C]
constexpr int OFF_K     = OFF_Q   + kNH * kN * kC;    // [NH, N, C]
constexpr int OFF_DIST  = OFF_K   + kNH * kN * kC;    // [E]
constexpr int OFF_LOGIT = OFF_DIST+ kE;               // [NH, E] (logit -> exp val)
constexpr int OFF_M     = OFF_LOGIT + kNH * kE;       // [NH, N] ordered-uint max
constexpr int OFF_Z     = OFF_M   + kNH * kN;         // [NH, N] softmax denom
constexpr int OFF_P     = OFF_Z   + kNH * kN;         // [NG, NH, C] pooled accum
constexpr int OFF_WVS   = OFF_P   + kNG * kNH * kC;   // [NH, E, C] scalar wv
constexpr int ZERO_BASE = OFF_M;
constexpr int ZERO_CNT  = OFF_WVS - OFF_M;            // m, z, P zeroed each launch

typedef __attribute__((ext_vector_type(2))) float v2f;
typedef __attribute__((ext_vector_type(8))) float v8f;

__device__ __forceinline__ v8f wmma_f32(v2f a, v2f b, v8f c) {
  // D = A(16x4) * B(4x16) + C, fp32 WMMA (CDNA5)
  return __builtin_amdgcn_wmma_f32_16x16x4_f32(false, a, false, b, (short)0, c,
                                               false, false);
}

// monotone float<->uint mapping for atomicMax on floats
__device__ __forceinline__ unsigned ford(float f) {
  unsigned u = __float_as_uint(f);
  return (u & 0x80000000u) ? ~u : (u | 0x80000000u);
}
__device__ __forceinline__ float forddec(unsigned u) {
  unsigned v = (u & 0x80000000u) ? (u & 0x7fffffffu) : ~u;
  return __uint_as_float(v);
}

// ---------------- kernel 0: zero m/z/P ----------------
__global__ void __launch_bounds__(256) init_kernel(float* __restrict__ ws) {
  int i = blockIdx.x * blockDim.x + threadIdx.x;
  if (i < ZERO_CNT) ws[ZERO_BASE + i] = 0.0f;  // bit pattern 0 == lowest ordered-uint
}

// ---------------- kernel 1: x = nf@We + b ; q,k = x@Wq/Wk (WMMA) ----------------
__global__ void __launch_bounds__(32) embed_qk_kernel(
    const float* __restrict__ nf, const float* __restrict__ We,
    const float* __restrict__ be, const float* __restrict__ Wq,
    const float* __restrict__ Wk, float* __restrict__ ws) {
  __shared__ float xt[16][17];
  const int node0 = blockIdx.x * 16;
  const int lane = threadIdx.x;
  const int col = lane & 15;   // row for A operand, col for B/C operands
  const int hi  = lane >> 4;
  float* xg = ws + OFF_X;
  float* qg = ws + OFF_Q;
  float* kg = ws + OFF_K;

  float bc = be[col];
  v8f xacc = {bc, bc, bc, bc, bc, bc, bc, bc};
#pragma unroll
  for (int kc = 0; kc < kFin / 4; ++kc) {
    const int kb = 4 * kc + 2 * hi;
    v2f a, b;
    a.x = nf[(node0 + col) * kFin + kb];
    a.y = nf[(node0 + col) * kFin + kb + 1];
    b.x = We[kb * kC + col];
    b.y = We[(kb + 1) * kC + col];
    xacc = wmma_f32(a, b, xacc);
  }
#pragma unroll
  for (int r = 0; r < 8; ++r) {
    const int m = r + 8 * hi;
    xt[m][col] = xacc[r];
    xg[(node0 + m) * kC + col] = xacc[r];
  }
  __syncthreads();
#pragma unroll
  for (int h = 0; h < kNH; ++h) {
    v8f qacc = {};
    v8f kacc = {};
#pragma unroll
    for (int kc = 0; kc < kC / 4; ++kc) {
      const int kb = 4 * kc + 2 * hi;
      v2f a, bq, bk;
      a.x = xt[col][kb];
      a.y = xt[col][kb + 1];
      bq.x = Wq[(h * kC + kb) * kC + col];
      bq.y = Wq[(h * kC + kb + 1) * kC + col];
      bk.x = Wk[(h * kC + kb) * kC + col];
      bk.y = Wk[(h * kC + kb + 1) * kC + col];
      qacc = wmma_f32(a, bq, qacc);
      kacc = wmma_f32(a, bk, kacc);
    }
#pragma unroll
    for (int r = 0; r < 8; ++r) {
      const int m = r + 8 * hi;
      qg[(h * kN + node0 + m) * kC + col] = qacc[r];
      kg[(h * kN + node0 + m) * kC + col] = kacc[r];
    }
  }
}

// ---------------- kernel 2: per-edge dist, logits, segment-max ----------------
__global__ void __launch_bounds__(256) edge_logit_kernel(
    const float* __restrict__ pos, const float* __restrict__ W1,
    const float* __restrict__ b1, const float* __restrict__ Wa,
    const int* __restrict__ ei, float* __restrict__ ws) {
  const int e = blockIdx.x * blockDim.x + threadIdx.x;
  if (e >= kE) return;
  const int s = ei[e];
  const int t = ei[kE + e];
  const float dx = pos[t * 3 + 0] - pos[s * 3 + 0];
  const float dy = pos[t * 3 + 1] - pos[s * 3 + 1];
  const float dz = pos[t * 3 + 2] - pos[s * 3 + 2];
  const float dist = sqrtf(dx * dx + dy * dy + dz * dz);
  ws[OFF_DIST + e] = dist;
  const float* qg = ws + OFF_Q;
  const float* kg = ws + OFF_K;
  unsigned* m_ord = (unsigned*)(ws + OFF_M);
#pragma unroll
  for (int h = 0; h < kNH; ++h) {
    const float df = (h == 0) ? dist : 1.0f / (dist * dist);
    float ab = 0.f;
#pragma unroll
    for (int j = 0; j < kHid; ++j) {
      const float hv = fmaxf(fmaf(W1[h * kHid + j], df, b1[h * kHid + j]), 0.f);
      ab = fmaf(hv, Wa[h * kHid + j], ab);
    }
    float dqk = 0.f;
#pragma unroll
    for (int c = 0; c < kC; ++c)
      dqk = fmaf(qg[(h * kN + t) * kC + c], kg[(h * kN + s) * kC + c], dqk);
    const float logit = dqk * 0.25f + ab;  // 1/sqrt(16)
    ws[OFF_LOGIT + h * kE + e] = logit;
    atomicMax(&m_ord[h * kN + t], ford(logit));
  }
}

// ------- kernel 3: wvs[h,e,c] = relu(W1*d+b1) @ Wv[:, 3c]  (WMMA, l=0 only) -------
__global__ void __launch_bounds__(32) wvs_kernel(
    const float* __restrict__ W1, const float* __restrict__ b1,
    const float* __restrict__ Wv, float* __restrict__ ws) {
  __shared__ float hl[16][68];
  const int tile = blockIdx.x;
  const int h  = tile / (kE / 16);
  const int e0 = (tile % (kE / 16)) * 16;
  const int lane = threadIdx.x;
  const int col = lane & 15;
  const int hi  = lane >> 4;
  const float dist = ws[OFF_DIST + e0 + col];
  const float df = (h == 0) ? dist : 1.0f / (dist * dist);
#pragma unroll
  for (int jj = 0; jj < 32; ++jj) {
    const int j = hi * 32 + jj;
    hl[col][j] = fmaxf(fmaf(W1[h * kHid + j], df, b1[h * kHid + j]), 0.f);
  }
  __syncthreads();
  v8f acc = {};
#pragma unroll
  for (int kc = 0; kc < kHid / 4; ++kc) {
    const int kb = 4 * kc + 2 * hi;
    v2f a, b;
    a.x = hl[col][kb];
    a.y = hl[col][kb + 1];
    b.x = Wv[(h * kHid + kb) * (kC * 3) + 3 * col];       // scalar-irrep column
    b.y = Wv[(h * kHid + kb + 1) * (kC * 3) + 3 * col];
    acc = wmma_f32(a, b, acc);
  }
  float* wvs = ws + OFF_WVS;
#pragma unroll
  for (int r = 0; r < 8; ++r) {
    const int m = r + 8 * hi;
    wvs[(h * kE + e0 + m) * kC + col] = acc[r];
  }
}

// ---------------- kernel 4: e = exp(logit - m[tgt]); z += e ----------------
__global__ void __launch_bounds__(256) softmax_z_kernel(
    const int* __restrict__ ei, float* __restrict__ ws) {
  const int idx = blockIdx.x * blockDim.x + threadIdx.x;  // idx = h*E + e
  if (idx >= kNH * kE) return;
  const int e = idx % kE;
  const int h = idx / kE;
  const int t = ei[kE + e];
  const unsigned* m_ord = (const unsigned*)(ws + OFF_M);
  const float m = forddec(m_ord[h * kN + t]);
  const float ev = __expf(ws[OFF_LOGIT + idx] - m);
  ws[OFF_LOGIT + idx] = ev;
  atomicAdd(&ws[OFF_Z + h * kN + t], ev);
}

// ---- kernel 5: P[g,h,c] += alpha * x[src,c] * wvs[h,e,c]  (LDS-block reduced) ----
__global__ void __launch_bounds__(256) message_kernel(
    const int* __restrict__ ei, const int* __restrict__ batch,
    float* __restrict__ ws) {
  __shared__ float Pl[kNG * kNH * kC];
  const int tid = threadIdx.x;
  for (int i = tid; i < kNG * kNH * kC; i += 256) Pl[i] = 0.f;
  __syncthreads();
  const int idx = blockIdx.x * 256 + tid;  // exact grid: NH*E / 256
  const int e = idx % kE;
  const int h = idx / kE;
  const int s = ei[e];
  const int t = ei[kE + e];
  const float alpha = ws[OFF_LOGIT + idx] / ws[OFF_Z + h * kN + t];
  const int g = batch[t];
  const float* xrow = ws + OFF_X + s * kC;
  const float* wvrow = ws + OFF_WVS + (h * kE + e) * kC;
  float* pl = &Pl[(g * kNH + h) * kC];
#pragma unroll
  for (int c = 0; c < kC; ++c)
    atomicAdd(&pl[c], alpha * xrow[c] * wvrow[c]);
  __syncthreads();
  float* Pg = ws + OFF_P;
  for (int i = tid; i < kNG * kNH * kC; i += 256) atomicAdd(&Pg[i], Pl[i]);
}

// ---------------- kernel 6: out = (P . Wo[l=0]) @ Wproj + bproj ----------------
__global__ void __launch_bounds__(256) finalize_kernel(
    const float* __restrict__ Wo, const float* __restrict__ Wproj,
    const float* __restrict__ bproj, const float* __restrict__ ws,
    float* __restrict__ out) {
  const int tid = threadIdx.x;
  const float* Pg = ws + OFF_P;
  for (int i = tid; i < kNG * kFout; i += 256) {
    const int g = i / kFout;
    const int f = i % kFout;
    float acc = bproj[f];
#pragma unroll
    for (int h = 0; h < kNH; ++h) {
#pragma unroll
      for (int o = 0; o < kC2; ++o) {
        float tsum = 0.f;
#pragma unroll
        for (int c = 0; c < kC; ++c)
          tsum = fmaf(Pg[(g * kNH + h) * kC + c],
                      Wo[((h * 3 + 0) * kC + c) * kC2 + o], tsum);
        acc = fmaf(tsum, Wproj[(h * kC2 + o) * kFout + f], acc);
      }
    }
    out[i] = acc;
  }
}

extern "C" void kernel_launch(void* const* d_in, const int* in_sizes, int n_in,
                              void* d_out, int out_size, void* d_ws, size_t ws_size,
                              hipStream_t stream) {
  const float* nf    = (const float*)d_in[0];
  const float* pos   = (const float*)d_in[1];
  const float* We    = (const float*)d_in[2];
  const float* be    = (const float*)d_in[3];
  const float* Wq    = (const float*)d_in[4];
  const float* Wk    = (const float*)d_in[5];
  const float* W1    = (const float*)d_in[6];
  const float* b1    = (const float*)d_in[7];
  const float* Wa    = (const float*)d_in[8];
  const float* Wv    = (const float*)d_in[9];
  const float* Wo    = (const float*)d_in[10];
  const float* Wproj = (const float*)d_in[11];
  const float* bproj = (const float*)d_in[12];
  const int*   ei    = (const int*)d_in[13];
  const int*   batch = (const int*)d_in[14];
  float* ws  = (float*)d_ws;
  float* out = (float*)d_out;

  init_kernel<<<(ZERO_CNT + 255) / 256, 256, 0, stream>>>(ws);
  embed_qk_kernel<<<kN / 16, 32, 0, stream>>>(nf, We, be, Wq, Wk, ws);
  edge_logit_kernel<<<kE / 256, 256, 0, stream>>>(pos, W1, b1, Wa, ei, ws);
  wvs_kernel<<<kNH * (kE / 16), 32, 0, stream>>>(W1, b1, Wv, ws);
  softmax_z_kernel<<<(kNH * kE) / 256, 256, 0, stream>>>(ei, ws);
  message_kernel<<<(kNH * kE) / 256, 256, 0, stream>>>(ei, batch, ws);
  finalize_kernel<<<1, 256, 0, stream>>>(Wo, Wproj, bproj, ws, out);
}